// SWTDecomposition_20787641713186
// MI455X (gfx1250) — compile-verified
//
#include <hip/hip_runtime.h>
#include <stdint.h>

typedef float v2f __attribute__((ext_vector_type(2)));
typedef float v8f __attribute__((ext_vector_type(8)));

static constexpr int Tlen = 4096;      // time length (power of 2, wrap via mask)
static constexpr int TM   = Tlen - 1;
static constexpr int CHK  = 512;       // output samples per workgroup chunk
static constexpr int NROW = 16;        // signal rows per workgroup (WMMA N dim)
static constexpr int Pw   = 676;       // LDS pitch (floats) for work buffers (bank-friendly, 16B-aligned rows)
static constexpr int Ps   = 2052;      // LDS pitch (floats) for staging (CHK*4 + 4)
static constexpr int XW   = CHK + 128; // 640: x tile width incl. 64-halo each side

__device__ __forceinline__ uint32_t lds_off(const void* p) {
  // low 32 bits of a generic pointer into LDS == DS byte offset
  return (uint32_t)(uintptr_t)p;
}
__device__ __forceinline__ void async_ld_b128(uint32_t lds, const float* g) {
  asm volatile("global_load_async_to_lds_b128 %0, %1, off" :: "v"(lds), "v"(g) : "memory");
}
__device__ __forceinline__ void async_st_b128(float* g, uint32_t lds) {
  asm volatile("global_store_async_from_lds_b128 %0, %1, off" :: "v"(g), "v"(lds) : "memory");
}
__device__ __forceinline__ void wait_async0() {
  asm volatile("s_wait_asynccnt 0x0" ::: "memory");
}

__global__ __launch_bounds__(256) void swt_db4_l3_kernel(
    const float* __restrict__ x,
    const float* __restrict__ dlo,
    const float* __restrict__ dhi,
    float* __restrict__ out)
{
  __shared__ alignas(16) float bufX [NROW * Pw]; // x tile, coords: t - (t0-64)
  __shared__ alignas(16) float bufL1[NROW * Pw]; // lo1, coords: t - (t0-48), width 608
  __shared__ alignas(16) float bufL2[NROW * Pw]; // lo2, coords: t - (t0-32), width 576
  __shared__ alignas(16) float stage[NROW * Ps]; // [row][t*4 + ch], t rel. t0

  const int tid   = threadIdx.x;
  const int lane  = tid & 31;
  const int wv    = tid >> 5;                 // wave id 0..7
  const int chunk = blockIdx.x & 7;
  const int rowg  = blockIdx.x >> 3;
  const int t0    = chunk * CHK;
  const int row0  = rowg * NROW;

  // ---------------- async load of x tile (wrap handled here) ----------------
  // 16 rows x 640 floats = 2560 16B units
  for (int f = tid; f < NROW * (XW / 4); f += 256) {
    int rr = f / (XW / 4);
    int lx = (f - rr * (XW / 4)) * 4;
    uint32_t l = lds_off(&bufX[rr * Pw + lx]);
    const float* g = x + ((size_t)(row0 + rr)) * Tlen + ((t0 - 64 + lx) & TM);
    async_ld_b128(l, g);
  }

  // ---------------- banded A-matrix weights (16x4 f32 layout) ----------------
  // A VGPR layout: lanes 0-15 -> M=lane, K={0,1}; lanes 16-31 -> M=lane-16, K={2,3}
  const int mA = lane & 15;
  const int k0 = (lane >> 4) * 2;
  v2f wlo[6], whi[6];
#pragma unroll
  for (int s = 0; s < 6; ++s) {
    int r  = -4 + 4 * s;
    int j0 = mA - r + 4 - k0;       // tap index for k = k0
    int j1 = j0 - 1;                // tap index for k = k0+1
    wlo[s].x = (j0 >= 0 && j0 < 8) ? dlo[j0] : 0.0f;
    wlo[s].y = (j1 >= 0 && j1 < 8) ? dlo[j1] : 0.0f;
    whi[s].x = (j0 >= 0 && j0 < 8) ? dhi[j0] : 0.0f;
    whi[s].y = (j1 >= 0 && j1 < 8) ? dhi[j1] : 0.0f;
  }

  const int nB = lane & 15;          // B/D column = signal row within group
  const int mh = (lane >> 4) * 8;    // D row offset for this half-wave

  // 6-slide accumulation: D[m,n] += sum_k h[m-r+4-k] * src[n][base + d*(r+k)]
  auto chain = [&](const float* src, int base, int d, v8f& clo, v8f& chi) {
#pragma unroll
    for (int s = 0; s < 6; ++s) {
      int rk = (-4 + 4 * s) + k0;
      v2f b;
      b.x = src[nB * Pw + base + d * rk];
      b.y = src[nB * Pw + base + d * (rk + 1)];
      clo = __builtin_amdgcn_wmma_f32_16x16x4_f32(false, wlo[s], false, b,
                                                  (short)0, clo, false, false);
      chi = __builtin_amdgcn_wmma_f32_16x16x4_f32(false, whi[s], false, b,
                                                  (short)0, chi, false, false);
    }
  };

  wait_async0();
  __syncthreads();

  // ---------------- Level 1 (d=1): lo1 on [0,608), hi1 -> stage ch3 ----------------
  for (int bi = wv; bi < 38; bi += 8) {
    int tb = 16 * bi;                         // lo1-local block start
    v8f clo = {}, chi = {};
    chain(bufX, tb + 16, 1, clo, chi);        // x is offset +16 vs lo1 coords
#pragma unroll
    for (int i = 0; i < 8; ++i)
      bufL1[nB * Pw + tb + (i + mh)] = clo[i];
    // interior test is block-uniform: t = tb-48+m in [0,512) for ALL m iff bi in [3,35)
    if (bi >= 3 && bi < 35) {
      int tb0 = tb - 48 + mh;                 // rel. t0
#pragma unroll
      for (int i = 0; i < 8; ++i)
        stage[nB * Ps + 4 * (tb0 + i) + 3] = chi[i];
    }
  }
  __syncthreads();

  // ---------------- Level 2 (d=2): lo2 on [0,576), hi2 -> stage ch2 ----------------
  for (int bi = wv; bi < 36; bi += 8) {
    int ph = bi & 1, ii = bi >> 1;
    int tb = ph + 32 * ii;                    // lo2-local block start
    v8f clo = {}, chi = {};
    chain(bufL1, tb + 16, 2, clo, chi);       // lo1 is offset +16 vs lo2 coords
#pragma unroll
    for (int i = 0; i < 8; ++i)
      bufL2[nB * Pw + tb + 2 * (i + mh)] = clo[i];
    // block-uniform interior: t = tb-32+2m in [0,512) for ALL m iff ii in [1,17)
    if (ii >= 1 && ii < 17) {
      int tb0 = tb - 32 + 2 * mh;             // rel. t0
#pragma unroll
      for (int i = 0; i < 8; ++i)
        stage[nB * Ps + 4 * (tb0 + 2 * i) + 2] = chi[i];
    }
  }
  __syncthreads();

  // ---------------- Level 3 (d=4): lo3 -> stage ch0, hi3 -> stage ch1 ----------------
  for (int bi = wv; bi < 32; bi += 8) {
    int ph = bi & 3, ii = bi >> 2;
    int tb = ph + 64 * ii;                    // rel. t0
    v8f clo = {}, chi = {};
    chain(bufL2, tb + 32, 4, clo, chi);       // lo2 is offset +32 vs output coords
#pragma unroll
    for (int i = 0; i < 8; ++i) {
      int t = tb + 4 * (i + mh);
      stage[nB * Ps + 4 * t + 0] = clo[i];    // adjacent floats: compiler can fuse
      stage[nB * Ps + 4 * t + 1] = chi[i];    // into ds_store_b64
    }
  }
  __syncthreads();

  // ---------------- coalesced async store: stage -> out[row][t][4] ----------------
  for (int f = tid; f < NROW * CHK; f += 256) {
    int rr = f >> 9;                // f / CHK
    int t  = f & (CHK - 1);
    uint32_t l = lds_off(&stage[rr * Ps + 4 * t]);
    float* g = out + (((size_t)(row0 + rr)) * Tlen + (t0 + t)) * 4;
    async_st_b128(g, l);
  }
  wait_async0();
}

extern "C" void kernel_launch(void* const* d_in, const int* in_sizes, int n_in,
                              void* d_out, int out_size, void* d_ws, size_t ws_size,
                              hipStream_t stream) {
  const float* x   = (const float*)d_in[0];
  const float* dlo = (const float*)d_in[1];
  const float* dhi = (const float*)d_in[2];
  float* out = (float*)d_out;
  // 2048 rows / 16 per WG = 128 row-groups; 4096 / 512 = 8 chunks
  dim3 grid(128 * 8), block(256);
  hipLaunchKernelGGL(swt_db4_l3_kernel, grid, block, 0, stream, x, dlo, dhi, out);
}